// GRU_8667244003631
// MI455X (gfx1250) — compile-verified
//
#include <hip/hip_runtime.h>
#include <math.h>

// CDNA5 fp32 WMMA: D(16x16,f32) = A(16x4,f32) x B(4x16,f32) + C
// A/B per-lane = 2 VGPRs (lanes 0-15: K=0,1 ; lanes 16-31: K=2,3), C/D = 8 VGPRs.
typedef float v2f __attribute__((ext_vector_type(2)));
typedef float v8f __attribute__((ext_vector_type(8)));
typedef unsigned int u32x4 __attribute__((ext_vector_type(4)));
typedef int i32x4 __attribute__((ext_vector_type(4)));
typedef int i32x8 __attribute__((ext_vector_type(8)));

__device__ __forceinline__ v8f wmma4(v2f a, v2f b, v8f c) {
  // 8 args: (neg_a, A, neg_b, B, c_mod, C, reuse_a, reuse_b)
  return __builtin_amdgcn_wmma_f32_16x16x4_f32(false, a, false, b, (short)0, c,
                                               false, false);
}

__device__ __forceinline__ float sigmoidf_(float v) {
  return 1.0f / (1.0f + expf(-v));
}

// ---------------------------------------------------------------------------
// Kernel 1: input projection  px[dir][t][b][g] = sum_k x[(b,t)][k] * w[dir][g][k]
// grid = (6 Nblocks of 128, T=1024, 2 dirs * 8 btiles), block = 256 (8 waves).
//
// The 16 x K fp32 x-tile is DMA'd into LDS by the Tensor Data Mover in ONE
// tensor_load_to_lds. TDM pad_enable inserts 4 DWORDs per 128 (K=128) or
// 2 DWORDs per 256 (K=512), so the LDS row stride is K+4 floats: stride mod
// 64 banks == 4, which makes the dual-half-wave b64 A-fragment reads cover
// all 64 banks exactly once (conflict-free) with zero staging VALU/ds work.
// ---------------------------------------------------------------------------
__global__ __launch_bounds__(256) void gru_inproj_kernel(
    const float* __restrict__ x, int xRowStride, int xTStride,
    const float* __restrict__ w, float* __restrict__ px, int K) {
  __shared__ float xs[16 * 516];  // 16 rows, row stride K+4 (max K=512 -> 516)
  const int nblk = blockIdx.x;
  const int t = blockIdx.y;
  const int dir = blockIdx.z >> 3;
  const int btile = blockIdx.z & 7;
  const int tid = threadIdx.x;

  const float* xp =
      x + (size_t)t * xTStride + (size_t)(btile * 16) * xRowStride;

  if (tid < 32) {  // one wave issues the tensor DMA
    const unsigned long long ga = (unsigned long long)(uintptr_t)xp;
    const unsigned lds = (unsigned)(uintptr_t)(&xs[0]);  // low 32 bits of the LDS address
    const unsigned interval = (K == 512) ? 7u : 6u;  // pad every 256 / 128 DW
    const unsigned amountF = (K == 512) ? 1u : 3u;   // pad 2 / 4 DW (field=n-1)

    u32x4 g0;
    g0.x = 1u;                          // count=1 (valid user descriptor)
    g0.y = lds;                         // lds_addr (bytes)
    g0.z = (unsigned)ga;                // global_addr[31:0]
    g0.w = (unsigned)((ga >> 32) & 0x01FFFFFFu) | (2u << 30);  // addr[56:32]|type=2

    i32x8 g1;
    g1[0] = (int)((2u << 16)            // data_size = 4 bytes
                  | (1u << 20)          // pad_enable
                  | (interval << 22) | (amountF << 25));  // wg_mask[15:0]=0
    g1[1] = (int)(((unsigned)K & 0xFFFFu) << 16);  // tensor_dim0[15:0]
    g1[2] = (int)((((unsigned)K >> 16) & 0xFFFFu)  // tensor_dim0[31:16]
                  | (16u << 16));                  // tensor_dim1[15:0] = 16
    g1[3] = (int)(((unsigned)K & 0xFFFFu) << 16);  // dim1 hi=0 | tile_dim0=K
    g1[4] = 16;                                    // tile_dim1=16, tile_dim2=0
    g1[5] = xRowStride;                            // tensor_dim0_stride[31:0]
    g1[6] = 0;                                     // stride hi | dim1_stride lo
    g1[7] = 0;
    i32x4 gz4 = {0, 0, 0, 0};
    i32x8 gz8 = {0, 0, 0, 0, 0, 0, 0, 0};

    // amdgpu-toolchain (clang-23) 6-arg form:
    // (g0, g1, g2, g3, <trailing group>, cpol)
    __builtin_amdgcn_tensor_load_to_lds(g0, g1, gz4, gz4, gz8, 0);
    __builtin_amdgcn_s_wait_tensorcnt((short)0);  // TENSORcnt==0: LDS written
  }
  __syncthreads();

  const int wave = tid >> 5;
  const int lane = tid & 31;
  const int ln = lane & 15;
  const int hi = (lane >> 4) << 1;  // 0 for lanes 0-15, 2 for lanes 16-31
  const int g0c = nblk * 128 + wave * 16;
  const int rstride = K + 4;  // padded LDS row stride (floats)
  const float* wrow = w + (size_t)dir * 768 * K + (size_t)(g0c + ln) * K;

  v8f acc = {0.f, 0.f, 0.f, 0.f, 0.f, 0.f, 0.f, 0.f};
  const int nk = K >> 2;
  for (int kc = 0; kc < nk; ++kc) {
    const int kk = (kc << 2) + hi;
    const int ka = kk + ((kk >> 8) << 1);  // +2 per 256-DW pad chunk crossed
    v2f a;
    a.x = xs[ln * rstride + ka];
    a.y = xs[ln * rstride + ka + 1];
    const v2f b = *(const v2f*)(wrow + kk);
    acc = wmma4(a, b, acc);
  }

  float* prow =
      px + ((size_t)(dir * 1024 + t) * 128 + btile * 16) * 768 + g0c + ln;
  const int mo = (lane < 16) ? 0 : 8;
#pragma unroll
  for (int e = 0; e < 8; ++e) prow[(size_t)(e + mo) * 768] = acc[e];
}

// ---------------------------------------------------------------------------
// Kernel 2: bidirectional GRU recurrence for one layer.
// grid = 16 (dir*8 + btile), block = 512 (16 waves). Wave w owns hidden
// columns j = w*16 + (lane&15); its r/z/n output tiles are at g = j, 256+j,
// 512+j, so the gate math is lane-local. w_hh fragments (3 gates x 64
// K-chunks x float2 = 384 VGPRs) are preloaded into registers once; the
// hidden state lives in LDS (k-major, stride 17, conflict-free both ways).
// ---------------------------------------------------------------------------
__global__ __launch_bounds__(512) void gru_recur_kernel(
    const float* __restrict__ px, const float* __restrict__ w_hh,
    const float* __restrict__ b_ih, const float* __restrict__ b_hh,
    float* __restrict__ y_out, float* __restrict__ h_fin) {
  __shared__ float hs[256 * 17];
  const int dir = blockIdx.x >> 3;
  const int btile = blockIdx.x & 7;
  const int tid = threadIdx.x;
  const int wave = tid >> 5;
  const int lane = tid & 31;
  const int ln = lane & 15;
  const int hi = (lane >> 4) << 1;
  const int j = wave * 16 + ln;        // owned hidden column
  const int mo = (lane < 16) ? 0 : 8;  // C-fragment row offset

  const float br = b_ih[dir * 768 + j] + b_hh[dir * 768 + j];
  const float bz = b_ih[dir * 768 + 256 + j] + b_hh[dir * 768 + 256 + j];
  const float bni = b_ih[dir * 768 + 512 + j];
  const float bnh = b_hh[dir * 768 + 512 + j];

  // Preload w_hh B-fragments into registers (constant over all 1024 steps).
  v2f bf[3][64];
#pragma unroll
  for (int G = 0; G < 3; ++G) {
    const float* wr = w_hh + ((size_t)dir * 768 + G * 256 + j) * 256;
#pragma unroll
    for (int kc = 0; kc < 64; ++kc)
      bf[G][kc] = *(const v2f*)(wr + (kc << 2) + hi);
  }

  for (int i = tid; i < 256 * 17; i += 512) hs[i] = 0.f;
  float hold[8];
#pragma unroll
  for (int e = 0; e < 8; ++e) hold[e] = 0.f;
  __syncthreads();

  const size_t pxDir = (size_t)dir * 1024 * 128 * 768;
  for (int s = 0; s < 1024; ++s) {
    const int t = dir ? (1023 - s) : s;

    // ph = h @ w_hh^T for the 3 gates (K = 256 via 64 fp32 WMMAs each)
    v8f cr = {0.f, 0.f, 0.f, 0.f, 0.f, 0.f, 0.f, 0.f};
    v8f cz = {0.f, 0.f, 0.f, 0.f, 0.f, 0.f, 0.f, 0.f};
    v8f cn = {0.f, 0.f, 0.f, 0.f, 0.f, 0.f, 0.f, 0.f};
#pragma unroll
    for (int kc = 0; kc < 64; ++kc) {
      const int kk = (kc << 2) + hi;
      v2f a;
      a.x = hs[kk * 17 + ln];
      a.y = hs[(kk + 1) * 17 + ln];
      cr = wmma4(a, bf[0][kc], cr);
      cz = wmma4(a, bf[1][kc], cz);
      cn = wmma4(a, bf[2][kc], cn);
    }

    const float* pxt =
        px + pxDir + (size_t)t * 128 * 768 + (size_t)(btile * 16) * 768;
    // prefetch next step's px tile (HBM-resident; ~1 step of lead time)
    if (s < 1023) {
      const int t2 = dir ? (t - 1) : (t + 1);
      const float* pxn =
          px + pxDir + (size_t)t2 * 128 * 768 + (size_t)(btile * 16) * 768;
#pragma unroll
      for (int e = 0; e < 8; ++e) {
        const float* pb = pxn + (size_t)(e + mo) * 768 + j;
        __builtin_prefetch(pb, 0, 1);
        __builtin_prefetch(pb + 256, 0, 1);
        __builtin_prefetch(pb + 512, 0, 1);
      }
    }

    float hnew[8];
#pragma unroll
    for (int e = 0; e < 8; ++e) {
      const int m = e + mo;
      const float* pb = pxt + (size_t)m * 768 + j;
      const float xr = pb[0];
      const float xz = pb[256];
      const float xn = pb[512];
      const float r = sigmoidf_(xr + cr[e] + br);
      const float z = sigmoidf_(xz + cz[e] + bz);
      const float n = tanhf(xn + bni + r * (cn[e] + bnh));
      const float h = (1.0f - z) * n + z * hold[e];
      hold[e] = h;
      hnew[e] = h;
    }

    __syncthreads();  // all waves done reading hs for this step's GEMM
    if (y_out != nullptr) {
      float* yrow =
          y_out + ((size_t)t * 128 + btile * 16) * 512 + dir * 256 + j;
#pragma unroll
      for (int e = 0; e < 8; ++e) {
        hs[j * 17 + (e + mo)] = hnew[e];
        yrow[(size_t)(e + mo) * 512] = hnew[e];
      }
    } else {
#pragma unroll
      for (int e = 0; e < 8; ++e) hs[j * 17 + (e + mo)] = hnew[e];
    }
    __syncthreads();  // new h visible before next step
  }

  if (h_fin != nullptr) {
    float* hr = h_fin + ((size_t)dir * 128 + btile * 16) * 256 + j;
#pragma unroll
    for (int e = 0; e < 8; ++e) hr[(size_t)(e + mo) * 256] = hold[e];
  }
}

// ---------------------------------------------------------------------------
// Kernel 3: output head  out[b][o] = concat(hf,hb)[b] . w_out[o] + b_out[o]
// ---------------------------------------------------------------------------
__global__ __launch_bounds__(512) void gru_head_kernel(
    const float* __restrict__ h_fin, const float* __restrict__ w_out,
    const float* __restrict__ b_out, float* __restrict__ out) {
  const int tid = threadIdx.x;  // 512 = 128 b * 4 o
  const int b = tid >> 2;
  const int o = tid & 3;
  const float* hf = h_fin + (size_t)b * 256;          // dir 0 final h
  const float* hb = h_fin + (size_t)(128 + b) * 256;  // dir 1 final h
  const float* wr = w_out + (size_t)o * 512;
  float acc = b_out[o];
  for (int k = 0; k < 256; ++k) acc += hf[k] * wr[k];
  for (int k = 0; k < 256; ++k) acc += hb[k] * wr[256 + k];
  out[(size_t)b * 4 + o] = acc;
}

// ---------------------------------------------------------------------------
extern "C" void kernel_launch(void* const* d_in, const int* in_sizes, int n_in,
                              void* d_out, int out_size, void* d_ws,
                              size_t ws_size, hipStream_t stream) {
  const float* data = (const float*)d_in[0];     // [128,1024,128]
  const float* w_ih_l0 = (const float*)d_in[1];  // [2,768,128]
  const float* w_hh_l0 = (const float*)d_in[2];  // [2,768,256]
  const float* b_ih_l0 = (const float*)d_in[3];  // [2,768]
  const float* b_hh_l0 = (const float*)d_in[4];  // [2,768]
  const float* w_ih_l1 = (const float*)d_in[5];  // [2,768,512]
  const float* w_hh_l1 = (const float*)d_in[6];  // [2,768,256]
  const float* b_ih_l1 = (const float*)d_in[7];  // [2,768]
  const float* b_hh_l1 = (const float*)d_in[8];  // [2,768]
  const float* w_out = (const float*)d_in[9];    // [4,512]
  const float* b_out = (const float*)d_in[10];   // [4]
  float* out = (float*)d_out;                    // [128,4]

  // workspace carve-up (floats): px reused between layers
  const size_t PX_ELEMS = (size_t)2 * 1024 * 128 * 768;  // 805 MB
  const size_t Y0_ELEMS = (size_t)1024 * 128 * 512;      // 268 MB
  float* px = (float*)d_ws;
  float* y0 = px + PX_ELEMS;
  float* hfin = y0 + Y0_ELEMS;  // [2,128,256]

  const dim3 gProj(6, 1024, 16);

  // ---- layer 0 ----
  gru_inproj_kernel<<<gProj, 256, 0, stream>>>(
      data, /*rowStride=*/1024 * 128, /*tStride=*/128, w_ih_l0, px, 128);
  gru_recur_kernel<<<16, 512, 0, stream>>>(px, w_hh_l0, b_ih_l0, b_hh_l0,
                                           /*y_out=*/y0, /*h_fin=*/nullptr);

  // ---- layer 1 (input = y0 [t][b][512]) ----
  gru_inproj_kernel<<<gProj, 256, 0, stream>>>(
      y0, /*rowStride=*/512, /*tStride=*/128 * 512, w_ih_l1, px, 512);
  gru_recur_kernel<<<16, 512, 0, stream>>>(px, w_hh_l1, b_ih_l1, b_hh_l1,
                                           /*y_out=*/nullptr, /*h_fin=*/hfin);

  // ---- output head ----
  gru_head_kernel<<<1, 512, 0, stream>>>(hfin, w_out, b_out, out);
}